// TemporalMultiHeadAttention_953482740300
// MI455X (gfx1250) — compile-verified
//
#include <hip/hip_runtime.h>
#include <hip/hip_bf16.h>

#define HIDDEN   128
#define HEADS    4
#define HEAD_DIM 32
#define TIME_DIM 32
#define SLOPE    0.2f
#define ETILES   4                 // 16-edge tiles per block in the WMMA kernel

typedef float v2f __attribute__((ext_vector_type(2)));
typedef float v8f __attribute__((ext_vector_type(8)));

// ---- ordered-uint encoding so float max can use global_atomic_max_u32 ----
__device__ __forceinline__ unsigned fenc(float f) {
    unsigned u = __float_as_uint(f);
    return (u & 0x80000000u) ? ~u : (u | 0x80000000u);
}
__device__ __forceinline__ float fdec(unsigned u) {
    unsigned v = (u & 0x80000000u) ? (u & 0x7FFFFFFFu) : ~u;
    return __uint_as_float(v);
}

__device__ __forceinline__ void wave_reduce4(float p[4]) {
    #pragma unroll
    for (int off = 16; off > 0; off >>= 1) {
        #pragma unroll
        for (int h = 0; h < 4; ++h) p[h] += __shfl_down(p[h], off, 32);
    }
}

// K0: zero output, init segment-max encodings and denominators
__global__ __launch_bounds__(256) void k_init(float* __restrict__ out,
                                              unsigned* __restrict__ menc,
                                              float* __restrict__ denom, int N) {
    size_t idx = (size_t)blockIdx.x * blockDim.x + threadIdx.x;
    size_t total = (size_t)N * HIDDEN;
    if (idx < total) out[idx] = 0.0f;
    if (idx < (size_t)N * HEADS) { menc[idx] = 0u; denom[idx] = 0.0f; }
}

// K1: fold attention vectors into projections:
//   c[sel][h][j] = sum_d a[h][sel*32+d] * W[(h*32+d)*128 + j]   (sel 0=Q, 1=K)
__global__ __launch_bounds__(256) void k_fold(const float* __restrict__ Wq,
                                              const float* __restrict__ Wk,
                                              const float* __restrict__ a,
                                              float* __restrict__ c) {
    int idx = blockIdx.x * blockDim.x + threadIdx.x;     // 0..1023
    if (idx >= 2 * HEADS * HIDDEN) return;
    int sel = idx >> 9;            // 0 = Wq/a_q, 1 = Wk/a_k
    int h   = (idx >> 7) & 3;
    int j   = idx & 127;
    const float* W = sel ? Wk : Wq;
    float s = 0.0f;
    #pragma unroll
    for (int d = 0; d < HEAD_DIM; ++d)
        s += a[h * 96 + sel * HEAD_DIM + d] * W[(size_t)(h * HEAD_DIM + d) * HIDDEN + j];
    c[idx] = s;
}

// K2: per-node Q score  sv[n][h] = h_v[n] . c_q[h]   (one wave32 per node)
__global__ __launch_bounds__(256) void k_node_scores(const float* __restrict__ h_v,
                                                     const float* __restrict__ c,
                                                     float* __restrict__ sv, int N) {
    int n    = (int)(((size_t)blockIdx.x * blockDim.x + threadIdx.x) >> 5);
    int lane = threadIdx.x & 31;
    if (n >= N) return;
    float4 x = ((const float4*)(h_v + (size_t)n * HIDDEN))[lane];
    float p[4];
    #pragma unroll
    for (int h = 0; h < 4; ++h) {
        float4 y = ((const float4*)(c + h * HIDDEN))[lane];
        p[h] = x.x * y.x + x.y * y.y + x.z * y.z + x.w * y.w;
    }
    wave_reduce4(p);
    if (lane == 0) {
        #pragma unroll
        for (int h = 0; h < 4; ++h) sv[(size_t)n * 4 + h] = p[h];
    }
}

// K3: per-edge raw score + leaky relu + segment max (one wave32 per edge)
__global__ __launch_bounds__(256) void k_edge_scores(const float* __restrict__ h_u,
                                                     const float* __restrict__ c,
                                                     const float* __restrict__ sv,
                                                     const float* __restrict__ delta_t,
                                                     const float* __restrict__ a,
                                                     const float* __restrict__ time_w,
                                                     const float* __restrict__ time_b,
                                                     const long long* __restrict__ tgt_idx,
                                                     float* __restrict__ raw,
                                                     unsigned* __restrict__ menc, int E) {
    int e    = (int)(((size_t)blockIdx.x * blockDim.x + threadIdx.x) >> 5);
    int lane = threadIdx.x & 31;
    if (e >= E) return;
    float4 x  = ((const float4*)(h_u + (size_t)e * HIDDEN))[lane];
    float  dt = delta_t[e];
    float  phi = __cosf(dt * time_w[lane] + time_b[lane]);   // lane == time index
    float p[4];
    #pragma unroll
    for (int h = 0; h < 4; ++h) {
        float4 y = ((const float4*)(c + 512 + h * HIDDEN))[lane]; // c_k at +512
        p[h] = x.x * y.x + x.y * y.y + x.z * y.z + x.w * y.w
             + phi * a[h * 96 + 2 * HEAD_DIM + lane];
    }
    wave_reduce4(p);
    if (lane == 0) {
        int t = (int)tgt_idx[e];
        #pragma unroll
        for (int h = 0; h < 4; ++h) {
            float r = p[h] + sv[(size_t)t * 4 + h];
            r = (r > 0.0f) ? r : SLOPE * r;
            raw[(size_t)e * 4 + h] = r;
            atomicMax(&menc[(size_t)t * 4 + h], fenc(r));
        }
    }
}

// K4: expv = exp(raw - m[tgt]); denom += expv     (thread per (edge,head))
__global__ __launch_bounds__(256) void k_expv(const long long* __restrict__ tgt_idx,
                                              const unsigned* __restrict__ menc,
                                              float* __restrict__ raw,
                                              float* __restrict__ denom, int E) {
    int idx = (int)((size_t)blockIdx.x * blockDim.x + threadIdx.x);
    if (idx >= E * 4) return;
    int e = idx >> 2, h = idx & 3;
    int t = (int)tgt_idx[e];
    float m  = fdec(menc[(size_t)t * 4 + h]);
    float ex = __expf(raw[idx] - m);
    raw[idx] = ex;                      // in-place: buffer now holds expv
    atomicAdd(&denom[(size_t)t * 4 + h], ex);
}

// K5: alpha = expv/(denom+eps) in place; emit alpha.T output slice
__global__ __launch_bounds__(256) void k_alpha(const long long* __restrict__ tgt_idx,
                                               const float* __restrict__ denom,
                                               float* __restrict__ raw,
                                               float* __restrict__ out_alphaT, int E) {
    int idx = (int)((size_t)blockIdx.x * blockDim.x + threadIdx.x);
    if (idx >= E * 4) return;
    int e = idx >> 2, h = idx & 3;
    int t = (int)tgt_idx[e];
    float al = raw[idx] / (denom[(size_t)t * 4 + h] + 1e-12f);
    raw[idx] = al;                      // buffer now holds alpha
    out_alphaT[(size_t)h * E + e] = al; // alpha.T layout [HEADS, E]
}

// K6: V = messages @ Wv.T via V_WMMA_F32_16X16X4_F32, scale by alpha, scatter-add.
// Block = 256 threads = 8 waves; each wave owns one 16-wide N-tile.
// Wv B-fragments (64 VGPRs/wave) are loaded ONCE and reused across ETILES
// 16-edge tiles staged together in LDS — removes the per-tile Wv reload stream.
__global__ __launch_bounds__(256) void k_vproj_scatter(const float* __restrict__ messages,
                                                       const float* __restrict__ Wv,
                                                       const float* __restrict__ alpha,
                                                       const long long* __restrict__ tgt_idx,
                                                       float* __restrict__ out, int E) {
    // pad +4: row stride 132 floats (528B, 8B aligned); lanes 0-15 hit banks
    // col+4i, lanes 16-31 hit col+2+4i -> disjoint, conflict-free b64 DS reads
    __shared__ float sM[ETILES * 16][HIDDEN + 4];
    __shared__ float sAl[ETILES * 16][4];
    __shared__ int   sT[ETILES * 16];

    int e0  = blockIdx.x * (ETILES * 16);
    int tid = threadIdx.x;

    // stage ETILES*16 message rows as float4: 64*32 = 2048 float4 over 256 threads
    for (int i = tid; i < ETILES * 16 * (HIDDEN / 4); i += 256) {
        int m  = i >> 5;           // row (32 float4 per row)
        int k4 = i & 31;
        int e  = e0 + m;
        float4 vld = make_float4(0.f, 0.f, 0.f, 0.f);
        if (e < E) vld = ((const float4*)(messages + (size_t)e * HIDDEN))[k4];
        sM[m][k4 * 4 + 0] = vld.x; sM[m][k4 * 4 + 1] = vld.y;
        sM[m][k4 * 4 + 2] = vld.z; sM[m][k4 * 4 + 3] = vld.w;
    }
    {   // 256 threads == ETILES*16*4 (m,h) slots exactly
        int m = tid >> 2, h = tid & 3;
        int e = e0 + m;
        sAl[m][h] = (e < E) ? alpha[(size_t)e * 4 + h] : 0.0f;
        if (h == 0) sT[m] = (e < E) ? (int)tgt_idx[e] : -1;
    }
    __syncthreads();

    int wave = tid >> 5;          // N-tile id: columns [wave*16, wave*16+16)
    int lane = tid & 31;
    int n0   = wave * 16;
    int koff = (lane >> 4) * 2;   // lanes 0-15 -> K{0,1}; lanes 16-31 -> K{2,3}
    int ncol = lane & 15;

    // B-fragments for this wave's 16 columns, resident in registers
    // B[k][n] = Wv.T[k][col] = Wv[col][k]
    v2f wvf[HIDDEN / 4];
    const float* wvrow = Wv + (size_t)(n0 + ncol) * HIDDEN;
    #pragma unroll
    for (int k = 0; k < HIDDEN / 4; ++k) {
        wvf[k].x = wvrow[k * 4 + koff];
        wvf[k].y = wvrow[k * 4 + koff + 1];
    }

    // C/D layout: VGPR r, lanes 0-15 -> (M=r, N=lane); lanes 16-31 -> (M=r+8, N=lane-16)
    int col  = n0 + (lane & 15);
    int head = col >> 5;
    int mhi  = (lane >> 4) * 8;

    for (int t = 0; t < ETILES; ++t) {
        int mrow = t * 16 + (lane & 15);    // A-frag row for this edge tile
        v8f acc = {};
        #pragma unroll
        for (int k = 0; k < HIDDEN / 4; ++k) {
            v2f af;
            af.x = sM[mrow][k * 4 + koff];
            af.y = sM[mrow][k * 4 + koff + 1];
            acc = __builtin_amdgcn_wmma_f32_16x16x4_f32(
                      false, af, false, wvf[k], (short)0, acc, false, false);
        }
        #pragma unroll
        for (int r = 0; r < 8; ++r) {
            int m  = t * 16 + r + mhi;
            int tt = sT[m];
            if (tt >= 0) {
                float val = acc[r] * sAl[m][head];
                atomicAdd(&out[(size_t)tt * HIDDEN + col], val);
            }
        }
    }
}

extern "C" void kernel_launch(void* const* d_in, const int* in_sizes, int n_in,
                              void* d_out, int out_size, void* d_ws, size_t ws_size,
                              hipStream_t stream) {
    const float*     h_v      = (const float*)d_in[0];
    const float*     h_u      = (const float*)d_in[1];
    const float*     delta_t  = (const float*)d_in[2];
    const float*     messages = (const float*)d_in[3];
    const float*     Wq       = (const float*)d_in[4];
    const float*     Wk       = (const float*)d_in[5];
    const float*     Wv       = (const float*)d_in[6];
    const float*     a        = (const float*)d_in[7];
    const float*     time_w   = (const float*)d_in[8];
    const float*     time_b   = (const float*)d_in[9];
    const long long* eidx     = (const long long*)d_in[10];  // int64 (2,E); row 0 = tgt

    const int N = in_sizes[0] / HIDDEN;   // num nodes / targets
    const int E = in_sizes[1] / HIDDEN;   // num edges

    float* out_nodes  = (float*)d_out;                        // [N,128]
    float* out_alphaT = (float*)d_out + (size_t)N * HIDDEN;   // [4,E]

    // workspace layout (all 4-byte elements)
    float*    ws_c     = (float*)d_ws;                 // 1024: c_q (512) then c_k (512)
    float*    ws_sv    = ws_c + 1024;                  // N*4 node scores
    float*    ws_raw   = ws_sv + (size_t)N * 4;        // E*4: raw -> expv -> alpha
    unsigned* ws_menc  = (unsigned*)(ws_raw + (size_t)E * 4);  // N*4 encoded max
    float*    ws_denom = (float*)(ws_menc + (size_t)N * 4);    // N*4 softmax denom

    const int B = 256;
    size_t init_total = (size_t)N * HIDDEN;
    k_init<<<(int)((init_total + B - 1) / B), B, 0, stream>>>(out_nodes, ws_menc, ws_denom, N);

    k_fold<<<(2 * HEADS * HIDDEN + B - 1) / B, B, 0, stream>>>(Wq, Wk, a, ws_c);

    k_node_scores<<<(int)(((size_t)N * 32 + B - 1) / B), B, 0, stream>>>(h_v, ws_c, ws_sv, N);

    k_edge_scores<<<(int)(((size_t)E * 32 + B - 1) / B), B, 0, stream>>>(
        h_u, ws_c, ws_sv, delta_t, a, time_w, time_b, eidx, ws_raw, ws_menc, E);

    k_expv<<<(int)(((size_t)E * 4 + B - 1) / B), B, 0, stream>>>(eidx, ws_menc, ws_raw, ws_denom, E);

    k_alpha<<<(int)(((size_t)E * 4 + B - 1) / B), B, 0, stream>>>(eidx, ws_denom, ws_raw, out_alphaT, E);

    k_vproj_scatter<<<(E + ETILES * 16 - 1) / (ETILES * 16), B, 0, stream>>>(
        messages, Wv, ws_raw, eidx, out_nodes, E);
}